// RoIHeadTemplate_56650618634941
// MI455X (gfx1250) — compile-verified
//
#include <hip/hip_runtime.h>
#include <stdint.h>

typedef unsigned int u32;
typedef unsigned long long u64;
typedef __attribute__((ext_vector_type(4))) u32 u32x4;
typedef __attribute__((ext_vector_type(8))) int i32x8;
typedef __attribute__((ext_vector_type(4))) int i32x4;

static constexpr int kB     = 4;
static constexpr int kN     = 32768;     // points per batch
static constexpr int kPre   = 4096;      // NMS_PRE_MAXSIZE
static constexpr int kPost  = 512;       // NMS_POST_MAXSIZE
static constexpr int kWords = kPre / 32; // 128 suppression-mask words per row
static constexpr float kThresh = 0.7f;

// ---------------------------------------------------------------------------
// Tensor Data Mover helpers (CDNA5 ISA).
// D# group0: count=1 | lds_addr | global_addr | type=2.
// D# group1: data_size | tensor_dim0/1 | tile_dim0/1 | dim0 stride.
// Builtin arity differs across toolchains (clang-22: 5 args, clang-23: 6).
// ---------------------------------------------------------------------------
#if __has_builtin(__builtin_amdgcn_tensor_load_to_lds) &&   \
    __has_builtin(__builtin_amdgcn_tensor_store_from_lds) && \
    __has_builtin(__builtin_amdgcn_s_wait_tensorcnt)
#define USE_TDM 1
#else
#define USE_TDM 0
#endif

#if USE_TDM
__device__ __forceinline__ void tdm_make_desc(u32x4& g0, i32x8& g1, u32 lds_off,
                                              const void* gaddr, u32 dsz, u32 d0,
                                              u32 td0, u32 d1, u32 td1, u32 s0) {
  u64 ga = (u64)(uintptr_t)gaddr;
  g0.x = 1u;                                             // count=1 (user desc)
  g0.y = lds_off;                                        // LDS byte offset
  g0.z = (u32)ga;                                        // global addr lo32
  g0.w = (u32)((ga >> 32) & 0x1FFFFFFull) | (2u << 30);  // addr[56:32] | type=2
  g1[0] = (int)(dsz << 16);                              // data_size code
  g1[1] = (int)((d0 & 0xFFFFu) << 16);                   // tensor_dim0 lo16
  g1[2] = (int)((d0 >> 16) | ((d1 & 0xFFFFu) << 16));    // dim0 hi | dim1 lo
  g1[3] = (int)((d1 >> 16) | ((td0 & 0xFFFFu) << 16));   // dim1 hi | tile_dim0
  g1[4] = (int)(td1 & 0xFFFFu);                          // tile_dim1, tile_dim2=0
  g1[5] = (int)s0;                                       // dim0 stride lo32
  g1[6] = 0;
  g1[7] = 0;
}

__device__ __forceinline__ void tdm_load(u32 lds_off, const void* src, u32 dsz,
                                         u32 d0, u32 td0, u32 d1, u32 td1, u32 s0) {
  u32x4 g0; i32x8 g1;
  tdm_make_desc(g0, g1, lds_off, src, dsz, d0, td0, d1, td1, s0);
  i32x4 gz = {0, 0, 0, 0};
#if defined(__clang_major__) && __clang_major__ >= 23
  i32x8 gz8 = {0, 0, 0, 0, 0, 0, 0, 0};
  __builtin_amdgcn_tensor_load_to_lds(g0, g1, gz, gz, gz8, 0);
#else
  __builtin_amdgcn_tensor_load_to_lds(g0, g1, gz, gz, 0);
#endif
}

__device__ __forceinline__ void tdm_store(u32 lds_off, void* dst, u32 dsz,
                                          u32 d0, u32 td0, u32 d1, u32 td1, u32 s0) {
  u32x4 g0; i32x8 g1;
  tdm_make_desc(g0, g1, lds_off, dst, dsz, d0, td0, d1, td1, s0);
  i32x4 gz = {0, 0, 0, 0};
#if defined(__clang_major__) && __clang_major__ >= 23
  i32x8 gz8 = {0, 0, 0, 0, 0, 0, 0, 0};
  __builtin_amdgcn_tensor_store_from_lds(g0, g1, gz, gz, gz8, 0);
#else
  __builtin_amdgcn_tensor_store_from_lds(g0, g1, gz, gz, 0);
#endif
}

// Low 32 bits of a flat LDS pointer == wave-relative LDS byte offset.
__device__ __forceinline__ u32 lds_off_of(const void* p) {
  return (u32)(uintptr_t)p;
}
#endif  // USE_TDM

// ---------------------------------------------------------------------------
// Zero the output buffer (reference scatters into zeros).
// ---------------------------------------------------------------------------
__global__ void k_zero(float* __restrict__ out, int n) {
  int g = blockIdx.x * blockDim.x + threadIdx.x;
  if (g < n) out[g] = 0.0f;
}

// ---------------------------------------------------------------------------
// Per-point score=max3 / pack sort key: (score_bits << 32) | (~index).
// Scores are uniform [0,1) so nonneg float bits are monotonic as u32;
// ~index makes ties prefer the smaller index (matches jax top_k).
// ---------------------------------------------------------------------------
__global__ void k_score(const float* __restrict__ cls, u64* __restrict__ keys) {
  int g = blockIdx.x * blockDim.x + threadIdx.x;
  if (g >= kB * kN) return;
  const float* c = cls + (size_t)g * 3;
  float m = fmaxf(c[0], fmaxf(c[1], c[2]));
  u32 n = (u32)(g & (kN - 1));
  keys[g] = ((u64)__float_as_uint(m) << 32) | (u64)(0xFFFFFFFFu - n);
}

// ---------------------------------------------------------------------------
// Bitonic sort of a 4096-key chunk, descending, fully in LDS (32 KB).
// Keys staged in/out by the Tensor Data Mover (one DMA each way).
// grid = (8 chunks, B), block = 256.
// ---------------------------------------------------------------------------
__global__ void __launch_bounds__(256) k_sortchunk(u64* __restrict__ keys) {
  __shared__ u64 s[kPre];
  const int bb = blockIdx.y, cc = blockIdx.x, tid = threadIdx.x;
  u64* base = keys + (size_t)bb * kN + (size_t)cc * kPre;
#if USE_TDM
  if (tid == 0) {
    tdm_load(lds_off_of(s), base, 3u, kPre, kPre, 1u, 1u, kPre);
    __builtin_amdgcn_s_wait_tensorcnt(0);
  }
#else
  for (int t = tid; t < kPre; t += 256) s[t] = base[t];
#endif
  __syncthreads();
  for (int k = 2; k <= kPre; k <<= 1) {
    for (int j = k >> 1; j > 0; j >>= 1) {
      for (int t = tid; t < kPre / 2; t += 256) {
        int i = ((t & ~(j - 1)) << 1) | (t & (j - 1));
        int p = i | j;
        u64 a = s[i], b2 = s[p];
        bool desc = ((i & k) == 0);
        if (desc ? (a < b2) : (a > b2)) { s[i] = b2; s[p] = a; }
      }
      __syncthreads();
    }
  }
#if USE_TDM
  if (tid == 0) tdm_store(lds_off_of(s), base, 3u, kPre, kPre, 1u, 1u, kPre);
  // S_ENDPGM implies wait-idle; LDS is live until the workgroup retires.
#else
  for (int t = tid; t < kPre; t += 256) base[t] = s[t];
#endif
}

// ---------------------------------------------------------------------------
// Top-4096 merge of two adjacent sorted-descending 4096 lists.
// Butterfly: lower[i] = max(A[i], B[n-1-i]) holds exactly the top-4096 and is
// bitonic; then a 4096-wide bitonic merge sorts it descending. Both chunks
// are contiguous, so a single 64 KB TDM load stages them.
// grid = (numMerges, B), block = 256.
// ---------------------------------------------------------------------------
__global__ void __launch_bounds__(256) k_merge(const u64* __restrict__ in,
                                               u64* __restrict__ out) {
  __shared__ u64 s[2 * kPre];
  const int bb = blockIdx.y, m = blockIdx.x, tid = threadIdx.x;
  const u64* src = in + (size_t)bb * kN + (size_t)m * (2 * kPre);
#if USE_TDM
  if (tid == 0) {
    tdm_load(lds_off_of(s), src, 3u, 2 * kPre, 2 * kPre, 1u, 1u, 2 * kPre);
    __builtin_amdgcn_s_wait_tensorcnt(0);
  }
#else
  for (int t = tid; t < 2 * kPre; t += 256) s[t] = src[t];
#endif
  __syncthreads();
  // Butterfly pass: keep per-pair max in the lower half (races impossible:
  // writes touch only [0,kPre), reads of the upper half are never written).
  for (int t = tid; t < kPre; t += 256) {
    u64 a = s[t], b2 = s[2 * kPre - 1 - t];
    if (b2 > a) s[t] = b2;
  }
  __syncthreads();
  // Bitonic merge of the (bitonic) lower half, descending.
  for (int j = kPre / 2; j > 0; j >>= 1) {
    for (int t = tid; t < kPre / 2; t += 256) {
      int i = ((t & ~(j - 1)) << 1) | (t & (j - 1));
      int p = i | j;
      u64 a = s[i], b2 = s[p];
      if (a < b2) { s[i] = b2; s[p] = a; }
    }
    __syncthreads();
  }
  u64* o = out + (size_t)bb * kN + (size_t)m * kPre;
#if USE_TDM
  if (tid == 0) tdm_store(lds_off_of(s), o, 3u, kPre, kPre, 1u, 1u, kPre);
#else
  for (int t = tid; t < kPre; t += 256) o[t] = s[t];
#endif
}

// ---------------------------------------------------------------------------
// Gather top-4096 boxes; emit SoA (x1,x2,y1,y2) planes [B][4][4096] for the
// IoU kernel (TDM source), plus raw 7-float boxes, scores, labels.
// grid = (16, B), block = 256.
// ---------------------------------------------------------------------------
__global__ void k_gather(const u64* __restrict__ keys, const float* __restrict__ boxes,
                         const float* __restrict__ cls, float* __restrict__ soa4,
                         float* __restrict__ boxes7, float* __restrict__ tscore,
                         int* __restrict__ tlabel) {
  const int bb = blockIdx.y;
  const int t = blockIdx.x * blockDim.x + threadIdx.x;  // 0..4095
  u64 key = keys[(size_t)bb * kN + t];
  u32 idx = 0xFFFFFFFFu - (u32)key;
  float score = __uint_as_float((u32)(key >> 32));
  const float* bx = boxes + ((size_t)bb * kN + idx) * 7;
  float x = bx[0], y = bx[1], dx = bx[3], dy = bx[4];
  float* S = soa4 + (size_t)bb * 4 * kPre;
  S[0 * kPre + t] = x - 0.5f * dx;
  S[1 * kPre + t] = x + 0.5f * dx;
  S[2 * kPre + t] = y - 0.5f * dy;
  S[3 * kPre + t] = y + 0.5f * dy;
  float* o7 = boxes7 + ((size_t)bb * kPre + t) * 7;
#pragma unroll
  for (int c = 0; c < 7; ++c) o7[c] = bx[c];
  tscore[bb * kPre + t] = score;
  const float* cc = cls + ((size_t)bb * kN + idx) * 3;
  int lab = 0;
  float mv = cc[0];
  if (cc[1] > mv) { mv = cc[1]; lab = 1; }
  if (cc[2] > mv) { mv = cc[2]; lab = 2; }
  tlabel[bb * kPre + t] = lab;
}

// ---------------------------------------------------------------------------
// Build the suppression bitmask matrix: mask[b][i][w] bit l set iff
// IoU(i, j=w*32+l) > 0.7 and j > i.  One wave per row; the 32 lanes of the
// wave cover 32 consecutive j and a wave32 ballot forms the word directly.
// Words are latched per-lane (lane l keeps word jw0+l) so each group of 32
// words is written with a single coalesced 32-lane store.
// Box SoA staged in LDS by the Tensor Data Mover (64 KB tile).
// grid = (4096/64, B), block = 256 (8 waves x 8 rows each).
// ---------------------------------------------------------------------------
__global__ void __launch_bounds__(256) k_mask(const float* __restrict__ soa4,
                                              u32* __restrict__ mask) {
  __shared__ float sb[4][kPre];  // 64 KB: x1,x2,y1,y2 planes
  const int bb = blockIdx.y;
  const float* src = soa4 + (size_t)bb * 4 * kPre;
#if USE_TDM
  if (threadIdx.x == 0) {
    tdm_load(lds_off_of(&sb[0][0]), src, 2u, kPre, kPre, 4u, 4u, kPre);
    __builtin_amdgcn_s_wait_tensorcnt(0);
  }
#else
  for (int t = threadIdx.x; t < 4 * kPre; t += 256) ((float*)sb)[t] = src[t];
#endif
  __syncthreads();

  const int wv = threadIdx.x >> 5;
  const int lane = threadIdx.x & 31;
  for (int r = 0; r < 8; ++r) {
    const int i = blockIdx.x * 64 + wv * 8 + r;
    const float x1i = sb[0][i], x2i = sb[1][i];
    const float y1i = sb[2][i], y2i = sb[3][i];
    const float ai = (x2i - x1i) * (y2i - y1i);
    u32* row = mask + ((size_t)bb * kPre + i) * kWords;
    for (int jw0 = 0; jw0 < kWords; jw0 += 32) {
      u32 myword = 0;
      for (int k = 0; k < 32; ++k) {
        const int j = (jw0 + k) * 32 + lane;
        const float x1j = sb[0][j], x2j = sb[1][j];
        const float y1j = sb[2][j], y2j = sb[3][j];
        float ix = fmaxf(fminf(x2i, x2j) - fmaxf(x1i, x1j), 0.0f);
        float iy = fmaxf(fminf(y2i, y2j) - fmaxf(y1i, y1j), 0.0f);
        float inter = ix * iy;
        float aj = (x2j - x1j) * (y2j - y1j);
        bool sup = (inter > kThresh * (ai + aj - inter + 1e-6f)) && (j > i);
#if __has_builtin(__builtin_amdgcn_ballot_w32)
        u32 word = __builtin_amdgcn_ballot_w32(sup);
#else
        u32 word = (u32)__ballot(sup);
#endif
        if (k == lane) myword = word;  // latch word jw0+lane in lane `lane`
      }
      row[jw0 + lane] = myword;  // coalesced 32-wide store
    }
  }
}

// ---------------------------------------------------------------------------
// Serial greedy NMS scan: one wave per batch. 4096-bit keep mask lives in
// VGPRs (4 words per lane); per step, broadcast the gate bit with __shfl and
// AND-out row i. No LDS / no barriers on the critical path.
// grid = B, block = 32.
// ---------------------------------------------------------------------------
__global__ void __launch_bounds__(32) k_scan(const u32* __restrict__ mask,
                                             u32* __restrict__ keep) {
  const int bb = blockIdx.x;
  const int l = threadIdx.x;
  const u32* M = mask + (size_t)bb * kPre * kWords + l;
  u32 kr0 = ~0u, kr1 = ~0u, kr2 = ~0u, kr3 = ~0u;
#pragma unroll 1
  for (int ph = 0; ph < 4; ++ph) {
#pragma unroll 4
    for (int ii = 0; ii < 1024; ++ii) {
      const int i = (ph << 10) + ii;
      u32 sel = (ph == 0) ? kr0 : (ph == 1) ? kr1 : (ph == 2) ? kr2 : kr3;
      u32 kw = (u32)__shfl((int)sel, (i >> 5) & 31, 32);
      u32 gate = 0u - ((kw >> (i & 31)) & 1u);  // all-ones iff box i kept
      const u32* row = M + (size_t)i * kWords;
      kr0 &= ~(row[0]  & gate);
      kr1 &= ~(row[32] & gate);
      kr2 &= ~(row[64] & gate);
      kr3 &= ~(row[96] & gate);
    }
  }
  keep[bb * kWords + l]      = kr0;
  keep[bb * kWords + l + 32] = kr1;
  keep[bb * kWords + l + 64] = kr2;
  keep[bb * kWords + l + 96] = kr3;
}

// ---------------------------------------------------------------------------
// Compact kept boxes (rank < 512) into the outputs. grid = B, block = 128
// (one thread per 32-bit keep word), simple LDS scan for global ranks.
// ---------------------------------------------------------------------------
__global__ void __launch_bounds__(128) k_compact(const u32* __restrict__ keep,
                                                 const float* __restrict__ boxes7,
                                                 const float* __restrict__ tscore,
                                                 const int* __restrict__ tlabel,
                                                 float* __restrict__ rois,
                                                 float* __restrict__ oscores,
                                                 float* __restrict__ olabels) {
  const int bb = blockIdx.x;
  const int t = threadIdx.x;  // word index 0..127
  __shared__ u32 pref[kWords];
  u32 kw = keep[bb * kWords + t];
  pref[t] = (u32)__popc(kw);
  __syncthreads();
  for (int off = 1; off < kWords; off <<= 1) {
    u32 v = (t >= off) ? pref[t - off] : 0u;
    __syncthreads();
    pref[t] += v;
    __syncthreads();
  }
  u32 r = pref[t] - (u32)__popc(kw);  // exclusive prefix = rank of first bit
  for (int bit = 0; bit < 32; ++bit) {
    if ((kw >> bit) & 1u) {
      if (r < (u32)kPost) {
        const int j = t * 32 + bit;
        const float* src = boxes7 + ((size_t)bb * kPre + j) * 7;
        float* dst = rois + ((size_t)bb * kPost + r) * 7;
#pragma unroll
        for (int c = 0; c < 7; ++c) dst[c] = src[c];
        oscores[bb * kPost + r] = tscore[bb * kPre + j];
        olabels[bb * kPost + r] = (float)(tlabel[bb * kPre + j] + 1);
      }
      ++r;
    }
  }
}

// ---------------------------------------------------------------------------
extern "C" void kernel_launch(void* const* d_in, const int* in_sizes, int n_in,
                              void* d_out, int out_size, void* d_ws, size_t ws_size,
                              hipStream_t stream) {
  (void)in_sizes; (void)n_in; (void)out_size; (void)ws_size;
  const float* boxes = (const float*)d_in[0];  // [B][N][7]
  const float* cls   = (const float*)d_in[1];  // [B][N][3]

  char* ws = (char*)d_ws;
  size_t off = 0;
  auto alloc = [&](size_t bytes) -> void* {
    void* p = ws + off;
    off = (off + bytes + 255) & ~(size_t)255;
    return p;
  };
  u64*   keysA   = (u64*)alloc((size_t)kB * kN * sizeof(u64));      // 1 MB
  u64*   keysB   = (u64*)alloc((size_t)kB * kN * sizeof(u64));      // 1 MB
  float* soa4    = (float*)alloc((size_t)kB * 4 * kPre * 4);        // 256 KB
  float* boxes7  = (float*)alloc((size_t)kB * kPre * 7 * 4);        // 448 KB
  float* tscore  = (float*)alloc((size_t)kB * kPre * 4);            // 64 KB
  int*   tlabel  = (int*)alloc((size_t)kB * kPre * 4);              // 64 KB
  u32*   maskbuf = (u32*)alloc((size_t)kB * kPre * kWords * 4);     // 8 MB
  u32*   keepbuf = (u32*)alloc((size_t)kB * kWords * 4);            // 2 KB

  float* rois    = (float*)d_out;
  float* oscores = rois + (size_t)kB * kPost * 7;
  float* olabels = oscores + (size_t)kB * kPost;
  const int nOut = kB * kPost * 9;  // 7 + 1 + 1 per slot

  k_zero<<<(nOut + 255) / 256, 256, 0, stream>>>((float*)d_out, nOut);
  k_score<<<(kB * kN + 255) / 256, 256, 0, stream>>>(cls, keysA);
  k_sortchunk<<<dim3(8, kB), 256, 0, stream>>>(keysA);
  k_merge<<<dim3(4, kB), 256, 0, stream>>>(keysA, keysB);   // 8 -> 4 chunks
  k_merge<<<dim3(2, kB), 256, 0, stream>>>(keysB, keysA);   // 4 -> 2 chunks
  k_merge<<<dim3(1, kB), 256, 0, stream>>>(keysA, keysB);   // 2 -> 1 (top 4096)
  k_gather<<<dim3(kPre / 256, kB), 256, 0, stream>>>(keysB, boxes, cls, soa4,
                                                     boxes7, tscore, tlabel);
  k_mask<<<dim3(kPre / 64, kB), 256, 0, stream>>>(soa4, maskbuf);
  k_scan<<<kB, 32, 0, stream>>>(maskbuf, keepbuf);
  k_compact<<<kB, 128, 0, stream>>>(keepbuf, boxes7, tscore, tlabel,
                                    rois, oscores, olabels);
}